// ICM_77403900609153
// MI455X (gfx1250) — compile-verified
//
#include <hip/hip_runtime.h>
#include <hip/hip_bf16.h>
#include <math.h>

#define N_NODES_C  50000
#define N_EDGES_C  800000
#define NDIM_C     256
#define HIDDEN_C   256
#define N_ACT_C    16
#define N_GRAPHS_C 64

typedef __attribute__((ext_vector_type(2))) float v2f;
typedef __attribute__((ext_vector_type(8))) float v8f;

// ---------------------------------------------------------------------------
// Degree / normalization: deg[i] = 1 (self loop) + #edges with dst==i
// dinv[i] = rsqrt(deg[i])   (deg >= 1 always, so no zero branch needed)
// ---------------------------------------------------------------------------
__global__ void k_deg_init(float* __restrict__ deg, int n) {
    int i = blockIdx.x * blockDim.x + threadIdx.x;
    if (i < n) deg[i] = 1.0f;
}

__global__ void k_deg_count(float* __restrict__ deg, const int* __restrict__ dst, int e) {
    int i = blockIdx.x * blockDim.x + threadIdx.x;
    if (i < e) atomicAdd(&deg[dst[i]], 1.0f);
}

__global__ void k_rsqrt(float* __restrict__ d, int n) {
    int i = blockIdx.x * blockDim.x + threadIdx.x;
    if (i < n) d[i] = rsqrtf(d[i]);
}

// ---------------------------------------------------------------------------
// Interleave weight rows in K pairs so a B fragment is one b64 load per lane:
//   Wp[p*512 + c*2 + 0] = W[2p  ][c]
//   Wp[p*512 + c*2 + 1] = W[2p+1][c]
// ---------------------------------------------------------------------------
__global__ __launch_bounds__(256) void k_ilw(const float* __restrict__ W,
                                             float* __restrict__ Wp) {
    int i = blockIdx.x * blockDim.x + threadIdx.x;   // 0 .. 32767 = p*256 + c
    int p = i >> 8;
    int c = i & 255;
    Wp[(size_t)p * 512 + c * 2]     = W[(size_t)(2 * p) * NDIM_C + c];
    Wp[(size_t)p * 512 + c * 2 + 1] = W[(size_t)(2 * p + 1) * NDIM_C + c];
}

// ---------------------------------------------------------------------------
// WMMA fp32 GEMM:  HS[m][n] = ACC[m][n] = dinv[m] * sum_k X[m][k] * W[k][n]
// M = 50000 (3125 tiles of 16), K = N = 256.
// Block = 256 threads = 8 wave32; one 16-row M tile per block, each wave
// computes two 16-wide N tiles with V_WMMA_F32_16X16X4_F32.
// Dual-destination epilogue seeds the aggregation accumulator (self-loop),
// removing the 51 MB D2D copy. ACC may alias the input X: each block stages
// its own 16 rows into LDS before any global store.
// ---------------------------------------------------------------------------
#define XS_STRIDE 260  // 256 + 4 pad: consecutive rows land on distinct banks

__global__ __launch_bounds__(256) void k_gemm(const float* __restrict__ X,
                                              const float* __restrict__ Wp,
                                              const float* __restrict__ dinv,
                                              float* __restrict__ HS,
                                              float* __restrict__ ACC) {
    __shared__ float xs[16 * XS_STRIDE];

    const int mbase = blockIdx.x * 16;
    const int tid   = threadIdx.x;

    // Cooperative coalesced load of the 16x256 A tile (float4 per thread x4).
    for (int i = tid; i < 16 * 64; i += 256) {
        int r  = i >> 6;
        int c4 = i & 63;
        const float4 v = reinterpret_cast<const float4*>(
            X + (size_t)(mbase + r) * NDIM_C)[c4];
        float* p = &xs[r * XS_STRIDE + c4 * 4];
        p[0] = v.x; p[1] = v.y; p[2] = v.z; p[3] = v.w;
    }
    __syncthreads();

    const int wave = tid >> 5;        // 0..7
    const int lane = tid & 31;
    const int lhi  = lane >> 4;       // 0: lanes 0-15, 1: lanes 16-31
    const int lr   = lane & 15;
    const int n0   = wave * 32;       // this wave owns columns [n0, n0+32)

    v8f acc0 = {0.f, 0.f, 0.f, 0.f, 0.f, 0.f, 0.f, 0.f};
    v8f acc1 = {0.f, 0.f, 0.f, 0.f, 0.f, 0.f, 0.f, 0.f};

    for (int k0 = 0; k0 < NDIM_C; k0 += 4) {
        // A fragment (16x4): lane holds row lr, K = k0 + 2*lhi + {0,1}
        const float* ap = &xs[lr * XS_STRIDE + k0 + 2 * lhi];
        v2f a; a.x = ap[0]; a.y = ap[1];

        // B fragments (4x16): K pair (k0 + 2*lhi, +1) interleaved -> one b64
        const float* wrow = Wp + (size_t)(k0 / 2 + lhi) * 512;
        v2f b0 = *reinterpret_cast<const v2f*>(wrow + (n0 + lr) * 2);
        v2f b1 = *reinterpret_cast<const v2f*>(wrow + (n0 + 16 + lr) * 2);

        acc0 = __builtin_amdgcn_wmma_f32_16x16x4_f32(false, a, false, b0,
                                                     (short)0, acc0, false, false);
        acc1 = __builtin_amdgcn_wmma_f32_16x16x4_f32(false, a, false, b1,
                                                     (short)0, acc1, false, false);
    }

    // C/D layout: VGPR r -> row (r + 8*lhi), col = lr within the N tile.
    #pragma unroll
    for (int r = 0; r < 8; ++r) {
        const int row = mbase + r + 8 * lhi;
        const float s = dinv[row];
        const float v0 = acc0[r] * s;
        const float v1 = acc1[r] * s;
        HS [(size_t)row * NDIM_C + n0 + lr]      = v0;
        HS [(size_t)row * NDIM_C + n0 + 16 + lr] = v1;
        ACC[(size_t)row * NDIM_C + n0 + lr]      = v0;
        ACC[(size_t)row * NDIM_C + n0 + 16 + lr] = v1;
    }
}

// ---------------------------------------------------------------------------
// Edge scatter-add: ACC[dst] += HS[src]   (one block of 256 per edge; the
// self-loop term was pre-seeded by the GEMM epilogue)
// ---------------------------------------------------------------------------
__global__ __launch_bounds__(256) void k_scatter(const float* __restrict__ hs,
                                                 float* __restrict__ acc,
                                                 const int* __restrict__ src,
                                                 const int* __restrict__ dst) {
    const int e = blockIdx.x;
    const int f = threadIdx.x;
    const int s = src[e];
    const int d = dst[e];
    atomicAdd(&acc[(size_t)d * NDIM_C + f], hs[(size_t)s * NDIM_C + f]);
}

// out = elu(dinv[i]*ACC[i] + b)
__global__ __launch_bounds__(256) void k_elu(const float* __restrict__ acc,
                                             const float* __restrict__ dinv,
                                             const float* __restrict__ b,
                                             float* __restrict__ out) {
    const int i = blockIdx.x;
    const int f = threadIdx.x;
    const float x = dinv[i] * acc[(size_t)i * NDIM_C + f] + b[f];
    out[(size_t)i * NDIM_C + f] = (x > 0.f) ? x : expm1f(x);
}

__global__ void k_zero(float* __restrict__ p, int n) {
    int i = blockIdx.x * blockDim.x + threadIdx.x;
    if (i < n) p[i] = 0.0f;
}

// pool[batch[i]] += dinv[i]*ACC[i] + b2 ; cnt[batch[i]] += 1
__global__ __launch_bounds__(256) void k_pool(const float* __restrict__ acc,
                                              const float* __restrict__ dinv,
                                              const float* __restrict__ b,
                                              const int* __restrict__ batch,
                                              float* __restrict__ pool,
                                              float* __restrict__ cnt) {
    const int i = blockIdx.x;
    const int f = threadIdx.x;
    const int g = batch[i];
    const float x = dinv[i] * acc[(size_t)i * NDIM_C + f] + b[f];
    atomicAdd(&pool[g * HIDDEN_C + f], x);
    if (f == 0) atomicAdd(&cnt[g], 1.0f);
}

__global__ __launch_bounds__(256) void k_pool_fin(const float* __restrict__ pool,
                                                  const float* __restrict__ cnt,
                                                  float* __restrict__ fout) {
    const int g = blockIdx.x;
    const int j = threadIdx.x;
    float c = cnt[g];
    c = (c > 1.0f) ? c : 1.0f;
    fout[g * HIDDEN_C + j] = pool[g * HIDDEN_C + j] / c;
}

// ---------------------------------------------------------------------------
// ForwardModel: out[g][j] = bf[j] + Wf[action[g]][j] + sum_k fs[g][k]*Wf[16+k][j]
// ---------------------------------------------------------------------------
__global__ __launch_bounds__(256) void k_forward(const float* __restrict__ fs,
                                                 const float* __restrict__ Wf,
                                                 const float* __restrict__ bf,
                                                 const int* __restrict__ action,
                                                 float* __restrict__ out) {
    const int g = blockIdx.x;
    const int j = threadIdx.x;
    float acc = bf[j] + Wf[(size_t)action[g] * HIDDEN_C + j];
    const float* fr = fs + (size_t)g * HIDDEN_C;
    for (int k = 0; k < HIDDEN_C; ++k)
        acc += fr[k] * Wf[(size_t)(N_ACT_C + k) * HIDDEN_C + j];
    out[(size_t)g * HIDDEN_C + j] = acc;
}

// ---------------------------------------------------------------------------
// InverseModel: z=[fs|fn]; relu(z@Wi1+b1); relu(@Wi2+b2); @Wi3+b3 -> [64,16]
// ---------------------------------------------------------------------------
__global__ __launch_bounds__(128) void k_inverse(const float* __restrict__ fs,
                                                 const float* __restrict__ fn,
                                                 const float* __restrict__ Wi1,
                                                 const float* __restrict__ bi1,
                                                 const float* __restrict__ Wi2,
                                                 const float* __restrict__ bi2,
                                                 const float* __restrict__ Wi3,
                                                 const float* __restrict__ bi3,
                                                 float* __restrict__ out) {
    const int g = blockIdx.x;
    const int t = threadIdx.x;  // 128
    __shared__ float zin[2 * HIDDEN_C];
    __shared__ float z1[128];
    __shared__ float z2[128];

    zin[t]       = fs[(size_t)g * HIDDEN_C + t];
    zin[t + 128] = fs[(size_t)g * HIDDEN_C + t + 128];
    zin[t + 256] = fn[(size_t)g * HIDDEN_C + t];
    zin[t + 384] = fn[(size_t)g * HIDDEN_C + t + 128];
    __syncthreads();

    float a = bi1[t];
    for (int k = 0; k < 2 * HIDDEN_C; ++k) a += zin[k] * Wi1[k * 128 + t];
    z1[t] = (a > 0.f) ? a : 0.f;
    __syncthreads();

    float a2 = bi2[t];
    for (int k = 0; k < 128; ++k) a2 += z1[k] * Wi2[k * 128 + t];
    z2[t] = (a2 > 0.f) ? a2 : 0.f;
    __syncthreads();

    if (t < N_ACT_C) {
        float a3 = bi3[t];
        for (int k = 0; k < 128; ++k) a3 += z2[k] * Wi3[k * N_ACT_C + t];
        out[(size_t)g * N_ACT_C + t] = a3;
    }
}

// ---------------------------------------------------------------------------
extern "C" void kernel_launch(void* const* d_in, const int* in_sizes, int n_in,
                              void* d_out, int out_size, void* d_ws, size_t ws_size,
                              hipStream_t stream) {
    (void)in_sizes; (void)n_in; (void)out_size; (void)ws_size;

    const float* x_s     = (const float*)d_in[0];
    const int*   ei_s    = (const int*)  d_in[1];
    const int*   batch_s = (const int*)  d_in[2];
    const float* x_n     = (const float*)d_in[3];
    const int*   ei_n    = (const int*)  d_in[4];
    const int*   batch_n = (const int*)  d_in[5];
    const int*   action  = (const int*)  d_in[6];
    const float* W1 = (const float*)d_in[7];   const float* b1 = (const float*)d_in[8];
    const float* W2 = (const float*)d_in[9];   const float* b2 = (const float*)d_in[10];
    const float* Wf = (const float*)d_in[11];  const float* bf = (const float*)d_in[12];
    const float* Wi1 = (const float*)d_in[13]; const float* bi1 = (const float*)d_in[14];
    const float* Wi2 = (const float*)d_in[15]; const float* bi2 = (const float*)d_in[16];
    const float* Wi3 = (const float*)d_in[17]; const float* bi3 = (const float*)d_in[18];

    float* out    = (float*)d_out;
    float* f_next = out;                                  // [64,256]
    float* nsh    = out + N_GRAPHS_C * HIDDEN_C;          // [64,256]
    float* ahat   = out + 2 * N_GRAPHS_C * HIDDEN_C;      // [64,16]

    const size_t NBIG = (size_t)N_NODES_C * NDIM_C;       // 12.8M floats
    float* bufA   = (float*)d_ws;
    float* bufB   = bufA + NBIG;
    float* dinv   = bufB + NBIG;
    float* fstate = dinv + N_NODES_C;
    float* pool   = fstate + N_GRAPHS_C * HIDDEN_C;
    float* cnt    = pool + N_GRAPHS_C * HIDDEN_C;         // contiguous after pool
    float* Wp1    = cnt + N_GRAPHS_C;                     // 65536 floats
    float* Wp2    = Wp1 + NDIM_C * HIDDEN_C;              // 65536 floats

    // K-pair interleave the two conv weights (once; reused by both graphs).
    k_ilw<<<(NDIM_C * HIDDEN_C / 2 + 255) / 256, 256, 0, stream>>>(W1, Wp1);
    k_ilw<<<(NDIM_C * HIDDEN_C / 2 + 255) / 256, 256, 0, stream>>>(W2, Wp2);

    auto run_graph = [&](const float* X, const int* ei, const int* batch,
                         float* ftarget) {
        const int* src = ei;
        const int* dst = ei + N_EDGES_C;

        k_deg_init<<<(N_NODES_C + 255) / 256, 256, 0, stream>>>(dinv, N_NODES_C);
        k_deg_count<<<(N_EDGES_C + 255) / 256, 256, 0, stream>>>(dinv, dst, N_EDGES_C);
        k_rsqrt<<<(N_NODES_C + 255) / 256, 256, 0, stream>>>(dinv, N_NODES_C);

        // conv1: HS1=bufA, ACC1=bufB seeded in epilogue; scatter; elu -> bufA
        k_gemm<<<N_NODES_C / 16, 256, 0, stream>>>(X, Wp1, dinv, bufA, bufB);
        k_scatter<<<N_EDGES_C, 256, 0, stream>>>(bufA, bufB, src, dst);
        k_elu<<<N_NODES_C, 256, 0, stream>>>(bufB, dinv, b1, bufA);

        // conv2: in=bufA, HS2=bufB, ACC2=bufA (safe: per-block LDS staging)
        k_gemm<<<N_NODES_C / 16, 256, 0, stream>>>(bufA, Wp2, dinv, bufB, bufA);
        k_scatter<<<N_EDGES_C, 256, 0, stream>>>(bufB, bufA, src, dst);

        k_zero<<<(N_GRAPHS_C * HIDDEN_C + N_GRAPHS_C + 255) / 256, 256, 0, stream>>>(
            pool, N_GRAPHS_C * HIDDEN_C + N_GRAPHS_C);
        k_pool<<<N_NODES_C, 256, 0, stream>>>(bufA, dinv, b2, batch, pool, cnt);
        k_pool_fin<<<N_GRAPHS_C, HIDDEN_C, 0, stream>>>(pool, cnt, ftarget);
    };

    run_graph(x_s, ei_s, batch_s, fstate);
    run_graph(x_n, ei_n, batch_n, f_next);

    k_forward<<<N_GRAPHS_C, HIDDEN_C, 0, stream>>>(fstate, Wf, bf, action, nsh);
    k_inverse<<<N_GRAPHS_C, 128, 0, stream>>>(fstate, f_next, Wi1, bi1, Wi2, bi2,
                                              Wi3, bi3, ahat);
}